// GCNConvModel_83339545411634
// MI455X (gfx1250) — compile-verified
//
#include <hip/hip_runtime.h>

typedef __attribute__((ext_vector_type(2))) float v2f;
typedef __attribute__((ext_vector_type(8))) float v8f;

// ---------------- utility: zero fill (grid-stride) ----------------
__global__ void zero_kernel(float* __restrict__ p, long n) {
    long i = (long)blockIdx.x * blockDim.x + threadIdx.x;
    long stride = (long)gridDim.x * blockDim.x;
    for (; i < n; i += stride) p[i] = 0.0f;
}

// ---------------- degree / dinv ----------------
__global__ void degree_kernel(const long long* __restrict__ col, unsigned* __restrict__ deg, int E) {
    int e = blockIdx.x * blockDim.x + threadIdx.x;
    if (e < E) atomicAdd(&deg[(int)col[e]], 1u);
}

__global__ void dinv_kernel(const unsigned* __restrict__ deg, float* __restrict__ dinv, int N) {
    int i = blockIdx.x * blockDim.x + threadIdx.x;
    if (i < N) dinv[i] = rsqrtf((float)deg[i] + 1.0f);
}

// ---------------- fp32 WMMA GEMM: out[M,N] = A[M,K] @ W[N,K]^T + bias ----------------
// Block = 256 threads (8 waves). Block owns 16 rows of A (staged in LDS),
// each wave produces 16x16 output tiles via V_WMMA_F32_16X16X4_F32.
// Requires: M % 16 == 0, N % 16 == 0, K % 4 == 0.
__global__ __launch_bounds__(256) void gemm_wmma_f32(
    const float* __restrict__ A, const float* __restrict__ W,
    const float* __restrict__ bias, float* __restrict__ out,
    int M, int N, int K, int ldo, int relu)
{
    extern __shared__ float lds[];                  // 16 * K floats
    const int m0  = blockIdx.x * 16;
    const int tid = threadIdx.x;

    // cooperative load of the 16xK A tile
    const int total = 16 * K;
    for (int i = tid; i < total; i += 256) {
        int r = i / K, c = i - r * K;
        lds[i] = A[(size_t)(m0 + r) * K + c];
    }
    __syncthreads();

    const int wave = tid >> 5;
    const int lane = tid & 31;
    const int hi   = lane >> 4;        // 0: K pair {k,k+1}; 1: K pair {k+2,k+3}
    const int lr   = lane & 15;
    const int ntiles = N >> 4;

    for (int nt = wave; nt < ntiles; nt += 8) {     // wave-uniform loop (EXEC stays full)
        const int n0   = nt << 4;
        const int ncol = n0 + lr;                   // D layout: N = lane & 15
        v8f c = {0.f, 0.f, 0.f, 0.f, 0.f, 0.f, 0.f, 0.f};

        for (int k = 0; k < K; k += 4) {
            const int bk = k + hi * 2;
            v2f a, b;
            // A-matrix 16x4 fp32 layout: lanes 0-15 M=0..15 K={k,k+1}; lanes 16-31 K={k+2,k+3}
            a[0] = lds[lr * K + bk];
            a[1] = lds[lr * K + bk + 1];
            // B-matrix 4x16 (KxN) = W^T chunk: B[kk][n] = W[n][kk]
            const float* wp = W + (size_t)ncol * K + bk;
            b[0] = wp[0];
            b[1] = wp[1];
            c = __builtin_amdgcn_wmma_f32_16x16x4_f32(
                    /*neg_a=*/false, a, /*neg_b=*/false, b,
                    /*c_mod=*/(short)0, c, /*reuse_a=*/false, /*reuse_b=*/false);
        }

        const float bv = bias ? bias[ncol] : 0.0f;
        #pragma unroll
        for (int r = 0; r < 8; ++r) {
            const int orow = m0 + r + hi * 8;       // D layout: lanes 16-31 hold M = r + 8
            float v = c[r] + bv;
            if (relu) v = fmaxf(v, 0.0f);
            out[(size_t)orow * ldo + ncol] = v;
        }
    }
}

// ---------------- edge scatter: agg[col] += h[row] * dinv[row]*dinv[col] ----------------
// One wave per edge; 32 lanes cover 128 features, 4 floats each (float4 gather, f32 atomics).
__global__ void edge_scatter(const float* __restrict__ h,
                             const long long* __restrict__ row,
                             const long long* __restrict__ col,
                             const float* __restrict__ dinv,
                             float* __restrict__ agg, int E)
{
    int gid = blockIdx.x * blockDim.x + threadIdx.x;
    int e = gid >> 5;
    if (e >= E) return;
    int f = (gid & 31) * 4;
    int r = (int)row[e];
    int c = (int)col[e];
    float en = dinv[r] * dinv[c];
    const float4 hv = *(const float4*)(h + (size_t)r * 128 + f);
    float* dst = agg + (size_t)c * 128 + f;
    atomicAdd(dst + 0, hv.x * en);
    atomicAdd(dst + 1, hv.y * en);
    atomicAdd(dst + 2, hv.z * en);
    atomicAdd(dst + 3, hv.w * en);
}

// ---------------- combine: h1 = relu(agg + dinv^2 * h + b) ----------------
__global__ void gcn_combine(const float* __restrict__ agg, const float* __restrict__ h,
                            const float* __restrict__ dinv, const float* __restrict__ b,
                            float* __restrict__ out, long total)
{
    long gid = (long)blockIdx.x * blockDim.x + threadIdx.x;
    if (gid >= total) return;
    int node = (int)(gid >> 7);
    int f = (int)(gid & 127);
    float sc = dinv[node];
    sc *= sc;
    out[gid] = fmaxf(agg[gid] + sc * h[gid] + b[f], 0.0f);
}

// ---------------- mean pool (sum + counts via atomics) ----------------
__global__ void pool_sum(const float* __restrict__ h, const long long* __restrict__ batch,
                         float* __restrict__ sums, float* __restrict__ counts, int Nn)
{
    int gid = blockIdx.x * blockDim.x + threadIdx.x;
    int node = gid >> 5;
    if (node >= Nn) return;
    int f = (gid & 31) * 4;
    int g = (int)batch[node];
    const float4 v = *(const float4*)(h + (size_t)node * 128 + f);
    float* dst = sums + (size_t)g * 128 + f;
    atomicAdd(dst + 0, v.x);
    atomicAdd(dst + 1, v.y);
    atomicAdd(dst + 2, v.z);
    atomicAdd(dst + 3, v.w);
    if ((gid & 31) == 0) atomicAdd(&counts[g], 1.0f);
}

// divide and write into concat buffer [G,192] columns 0..127
__global__ void pool_div_concat(const float* __restrict__ sums, const float* __restrict__ counts,
                                float* __restrict__ cat, int G)
{
    int gid = blockIdx.x * blockDim.x + threadIdx.x;
    if (gid >= G * 128) return;
    int g = gid >> 7, f = gid & 127;
    float cnt = fmaxf(counts[g], 1.0f);
    cat[(size_t)g * 192 + f] = sums[gid] / cnt;
}

// ---------------- final [G,256] @ out_W[1,256]^T + out_b: one wave per graph ----------------
__global__ void final_out(const float* __restrict__ h, const float* __restrict__ w,
                          const float* __restrict__ b, float* __restrict__ out, int G)
{
    int wv = (blockIdx.x * blockDim.x + threadIdx.x) >> 5;
    int lane = threadIdx.x & 31;
    if (wv >= G) return;
    float s = 0.0f;
    for (int k = lane; k < 256; k += 32) s += h[(size_t)wv * 256 + k] * w[k];
    #pragma unroll
    for (int off = 16; off > 0; off >>= 1) s += __shfl_down(s, off, 32);
    if (lane == 0) out[wv] = s + b[0];
}

extern "C" void kernel_launch(void* const* d_in, const int* in_sizes, int n_in,
                              void* d_out, int out_size, void* d_ws, size_t ws_size,
                              hipStream_t stream)
{
    const float*     x         = (const float*)d_in[0];
    const long long* ei        = (const long long*)d_in[1];
    const long long* batch     = (const long long*)d_in[2];
    const float*     mol       = (const float*)d_in[3];
    const float*     gcn_W     = (const float*)d_in[4];
    const float*     gcn_b     = (const float*)d_in[5];
    const float*     gcn_out_W = (const float*)d_in[6];
    const float*     gcn_out_b = (const float*)d_in[7];
    const float*     mlp_W     = (const float*)d_in[8];
    const float*     mlp_b     = (const float*)d_in[9];
    const float*     mlp_out_W = (const float*)d_in[10];
    const float*     mlp_out_b = (const float*)d_in[11];
    const float*     pred_W1   = (const float*)d_in[12];
    const float*     pred_b1   = (const float*)d_in[13];
    const float*     pred_W2   = (const float*)d_in[14];
    const float*     pred_b2   = (const float*)d_in[15];
    const float*     out_W     = (const float*)d_in[16];
    const float*     out_b     = (const float*)d_in[17];
    float* out = (float*)d_out;

    const int Nn = 50000, E = 800000, G = 256;
    const long long* row = ei;
    const long long* col = ei + E;

    // workspace carve-out (256B aligned)
    char* ws = (char*)d_ws;
    size_t off = 0;
    auto alloc = [&](size_t bytes) -> char* {
        off = (off + 255) & ~(size_t)255;
        char* p = ws + off;
        off += bytes;
        return p;
    };
    unsigned* deg  = (unsigned*)alloc((size_t)Nn * 4);
    float*    dinv = (float*)   alloc((size_t)Nn * 4);
    float*    h    = (float*)   alloc((size_t)Nn * 128 * 4);
    float*    h1   = (float*)   alloc((size_t)Nn * 128 * 4);
    float*    agg  = (float*)   alloc((size_t)Nn * 128 * 4);
    float*    sums = (float*)   alloc((size_t)G * 128 * 4);
    float*    cnts = (float*)   alloc((size_t)G * 4);
    float*    h2a  = (float*)   alloc((size_t)G * 256 * 4);
    float*    h2b  = (float*)   alloc((size_t)G * 256 * 4);
    float*    cat  = (float*)   alloc((size_t)G * 192 * 4);
    float*    p1   = (float*)   alloc((size_t)G * 256 * 4);
    float*    p2   = (float*)   alloc((size_t)G * 256 * 4);
    (void)ws_size; (void)in_sizes; (void)n_in; (void)out_size;

    // degrees (self-loop handled inside dinv via +1)
    zero_kernel<<<512, 256, 0, stream>>>((float*)deg, Nn);
    degree_kernel<<<(E + 255) / 256, 256, 0, stream>>>(col, deg, E);
    dinv_kernel<<<(Nn + 255) / 256, 256, 0, stream>>>(deg, dinv, Nn);

    const long nodeElems = (long)Nn * 128;

    // 3 GCN layers
    for (int i = 0; i < 3; ++i) {
        const float* A = (i == 0) ? x : h1;
        gemm_wmma_f32<<<Nn / 16, 256, 16 * 128 * 4, stream>>>(
            A, gcn_W + (size_t)i * 128 * 128, nullptr, h, Nn, 128, 128, 128, 0);
        zero_kernel<<<2048, 256, 0, stream>>>(agg, nodeElems);
        edge_scatter<<<(E * 32 + 255) / 256, 256, 0, stream>>>(h, row, col, dinv, agg, E);
        gcn_combine<<<(int)((nodeElems + 255) / 256), 256, 0, stream>>>(
            agg, h, dinv, gcn_b + i * 128, h1, nodeElems);
    }

    // final GCN-branch linear (no relu)
    gemm_wmma_f32<<<Nn / 16, 256, 16 * 128 * 4, stream>>>(
        h1, gcn_out_W, gcn_out_b, h, Nn, 128, 128, 128, 0);

    // mean pool -> concat[:, 0:128]
    zero_kernel<<<64, 256, 0, stream>>>(sums, (long)G * 128);
    zero_kernel<<<1, 256, 0, stream>>>(cnts, G);
    pool_sum<<<(Nn * 32 + 255) / 256, 256, 0, stream>>>(h, batch, sums, cnts, Nn);
    pool_div_concat<<<(G * 128 + 255) / 256, 256, 0, stream>>>(sums, cnts, cat, G);

    // MLP branch -> concat[:, 128:192]
    gemm_wmma_f32<<<G / 16, 256, 16 * 256 * 4, stream>>>(mol, mlp_W, mlp_b, h2a, G, 256, 256, 256, 1);
    gemm_wmma_f32<<<G / 16, 256, 16 * 256 * 4, stream>>>(h2a, mlp_W + 256 * 256, mlp_b + 256, h2b, G, 256, 256, 256, 1);
    gemm_wmma_f32<<<G / 16, 256, 16 * 256 * 4, stream>>>(h2b, mlp_out_W, mlp_out_b, cat + 128, G, 64, 256, 192, 1);

    // prediction head
    gemm_wmma_f32<<<G / 16, 256, 16 * 192 * 4, stream>>>(cat, pred_W1, pred_b1, p1, G, 256, 192, 256, 1);
    gemm_wmma_f32<<<G / 16, 256, 16 * 256 * 4, stream>>>(p1, pred_W2, pred_b2, p2, G, 256, 256, 256, 1);
    final_out<<<(G * 32 + 255) / 256, 256, 0, stream>>>(p2, out_W, out_b, out, G);
}